// MaxCrop_26688926777801
// MI455X (gfx1250) — compile-verified
//
#include <hip/hip_runtime.h>

typedef __attribute__((ext_vector_type(2))) float v2f;
typedef __attribute__((ext_vector_type(8))) float v8f;

#define NC    9
#define H     128
#define W     128
#define KWIN  102           // int(0.8 * 128): box kernel and crop size
#define OW    27            // 128 - 102 + 1
#define PLANE (H * W)       // 16384

__launch_bounds__(256, 4)
__global__ void maxcrop_kernel(const float* __restrict__ x, float* __restrict__ out) {
    const int n    = blockIdx.x;
    const int t    = threadIdx.x;
    const int lane = t & 31;
    const int wave = t >> 5;            // 0..7

    __shared__ float s[H * W];          // 64 KB: channel-summed image (later: column prefix)
    __shared__ float colwin[OW * W];    // 13.5 KB: 102-tall column windows (later: row prefix)
    __shared__ float bestVal[OW];
    __shared__ int   bestIdx[OW];
    __shared__ int   sPy, sPx;
    __shared__ int   rowMap[H];         // (py + mh[i]) * W
    __shared__ int   colMap[W];         // px + mw[j]

    const float* xn = x + (size_t)n * NC * PLANE;

    // ---- Step 1: channel sum via V_WMMA_F32_16X16X4_F32 ----
    // A = [16 pixels x 4 channels] f32, B = ones(4x16), chained C over 3 K-chunks.
    // A layout: lanes 0-15 hold M=0..15 (K0,K1 in v[0],v[1]); lanes 16-31 hold K2,K3.
    // Sum over K is invariant to K-slot permutation, so each lane loads 2 distinct channels.
    {
        const int half = lane >> 4;     // 0 -> channels {0,1},{4,5},{8}; 1 -> {2,3},{6,7}
        const int m    = lane & 15;
        v2f bones; bones[0] = 1.0f; bones[1] = 1.0f;

        for (int g = wave; g < PLANE / 16; g += 8) {
            const int pix = g * 16 + m;
            v2f a0, a1, a2;
            a0[0] = xn[(half * 2 + 0) * PLANE + pix];        // chunk 0: channels 0..3
            a0[1] = xn[(half * 2 + 1) * PLANE + pix];
            a1[0] = xn[(4 + half * 2 + 0) * PLANE + pix];    // chunk 1: channels 4..7
            a1[1] = xn[(4 + half * 2 + 1) * PLANE + pix];
            a2[0] = half ? 0.0f : xn[8 * PLANE + pix];       // chunk 2: channel 8 + zero pad
            a2[1] = 0.0f;

            v8f c = {};
            c = __builtin_amdgcn_wmma_f32_16x16x4_f32(false, a0, false, bones, (short)0, c, false, false);
            c = __builtin_amdgcn_wmma_f32_16x16x4_f32(false, a1, false, bones, (short)0, c, false, false);
            c = __builtin_amdgcn_wmma_f32_16x16x4_f32(false, a2, false, bones, (short)0, c, false, false);

            // D: VGPR r -> M=r (lanes 0-15) / M=r+8 (lanes 16-31); all N columns identical.
            if ((lane & 15) == 0) {
                float* dst = &s[g * 16 + half * 8];
                dst[0] = c[0]; dst[1] = c[1]; dst[2] = c[2]; dst[3] = c[3];
                dst[4] = c[4]; dst[5] = c[5]; dst[6] = c[6]; dst[7] = c[7];
            }
        }
    }
    __syncthreads();

    // ---- Step 2: column inclusive prefix sums (in place), then 102-tall windows ----
    if (t < W) {
        float run = 0.0f;
        for (int y = 0; y < H; ++y) { run += s[y * W + t]; s[y * W + t] = run; }
    }
    __syncthreads();

    for (int i = t; i < OW * W; i += 256) {
        const int y = i >> 7, xc = i & 127;
        colwin[i] = s[(y + KWIN - 1) * W + xc] - (y ? s[(y - 1) * W + xc] : 0.0f);
    }
    __syncthreads();

    // ---- Step 3: row prefix (in place) + 102-wide windows + per-row first-argmax ----
    if (t < OW) {
        float run = 0.0f;
        for (int xc = 0; xc < W; ++xc) { run += colwin[t * W + xc]; colwin[t * W + xc] = run; }
        float bv = -__builtin_inff();
        int   bx = 0;
        for (int xp = 0; xp < OW; ++xp) {
            const float v = colwin[t * W + xp + KWIN - 1] - (xp ? colwin[t * W + xp - 1] : 0.0f);
            if (v > bv) { bv = v; bx = xp; }    // strict > keeps first max in x
        }
        bestVal[t] = bv;
        bestIdx[t] = t * OW + bx;
    }
    __syncthreads();

    if (t == 0) {
        float bv = bestVal[0]; int bi = bestIdx[0];
        for (int y = 1; y < OW; ++y)
            if (bestVal[y] > bv) { bv = bestVal[y]; bi = bestIdx[y]; }  // keeps lowest row on tie
        sPy = bi / OW;
        sPx = bi % OW;
    }
    __syncthreads();

    // ---- Step 4: index maps + crop-resize gather (reads mostly hit 192MB L2) ----
    if (t < W) {
        colMap[t] = sPx + ((t * KWIN) >> 7);
        rowMap[t] = (sPy + ((t * KWIN) >> 7)) * W;
    }
    __syncthreads();

    float* outn = out + (size_t)n * NC * PLANE;
    for (int i = t; i < NC * PLANE; i += 256) {
        const int j  = i & 127;
        const int r  = (i >> 7) & 127;
        const int ch = i >> 14;
        outn[i] = xn[ch * PLANE + rowMap[r] + colMap[j]];   // coalesced store, banded gather read
    }
}

extern "C" void kernel_launch(void* const* d_in, const int* in_sizes, int n_in,
                              void* d_out, int out_size, void* d_ws, size_t ws_size,
                              hipStream_t stream) {
    const float* x  = (const float*)d_in[0];
    float* out      = (float*)d_out;
    const int nImgs = in_sizes[0] / (NC * PLANE);   // 256 for the reference setup
    maxcrop_kernel<<<nImgs, 256, 0, stream>>>(x, out);
}